// Pooling_block_53884659696259
// MI455X (gfx1250) — compile-verified
//
#include <hip/hip_runtime.h>

// Problem constants (fixed by the harness setup).
#define B_ 4
#define N_ 4096
#define C_ 96
#define PS 4                 // pooling_patch_size
#define NP (N_ / PS)         // 1024 patches, top-k with k = PS/reduction_ratio = 1
#define OUTROWS (NP + 1)     // 1025 (cls token + pooled)

typedef __attribute__((ext_vector_type(4))) unsigned int v4u;
typedef __attribute__((ext_vector_type(8))) int          v8i;
typedef __attribute__((ext_vector_type(4))) int          v4i;
typedef __attribute__((ext_vector_type(2))) float        v2f;
typedef __attribute__((ext_vector_type(8))) float        v8f;

#if defined(__has_builtin)
#if __has_builtin(__builtin_amdgcn_tensor_load_to_lds) && __has_builtin(__builtin_amdgcn_s_wait_tensorcnt)
#define USE_TDM 1
#endif
#if __has_builtin(__builtin_amdgcn_tanhf)
#define HAVE_TANH 1
#endif
#endif

__device__ __forceinline__ float sigmoidf_fast(float x) {
#ifdef HAVE_TANH
  // sigmoid(x) = 0.5*tanh(x/2) + 0.5  -> single v_tanh_f32 + fma
  return __builtin_fmaf(__builtin_amdgcn_tanhf(x * 0.5f), 0.5f, 0.5f);
#else
  // 1 / (1 + e^-x) = rcp(1 + exp2(-x*log2(e)))
  return __builtin_amdgcn_rcpf(1.0f + __builtin_amdgcn_exp2f(x * -1.44269504088896340736f));
#endif
}

__device__ __forceinline__ float waveAllSum(float v) {
#pragma unroll
  for (int off = 16; off >= 1; off >>= 1) v += __shfl_xor(v, off, 32);
  return v;
}

// ---------------------------------------------------------------------------
// K1: t[b,m] = sum_c sigmoid(x[b,1+m,c]) * theta[c].   One wave per node.
// ---------------------------------------------------------------------------
__global__ __launch_bounds__(256) void node_gate_kernel(
    const float* __restrict__ x, const float* __restrict__ theta,
    float* __restrict__ t) {
  const int wave = threadIdx.x >> 5;
  const int lane = threadIdx.x & 31;
  const int m = blockIdx.x * 8 + wave;  // 0 .. B*N-1
  const int b = m / N_;
  const int r = m - b * N_;
  const float* row = x + ((size_t)b * (N_ + 1) + 1 + r) * C_;
  float acc = 0.f;
#pragma unroll
  for (int j = 0; j < 3; ++j) {        // C = 96 = 3 * 32 lanes
    const int c = lane + 32 * j;
    acc += sigmoidf_fast(row[c]) * theta[c];
  }
  acc = waveAllSum(acc);
  if (lane == 0) t[m] = acc;
}

// ---------------------------------------------------------------------------
// K2: scores[b,n] = sum_m sigmoid(edge[b,n,m]) * t[b,m]  via f32 WMMA.
//
// Block = 16 rows x K=4096.  8 waves each own a 512-wide K-chunk and run
// D(16x16) += A(16x4) * B(4x16) with every column of B equal to t[k..k+3];
// column 0 of D then holds the 16 row-scores.  A's ISA layout (lane L:
// row L&15, K-pair 2*(L>>4)) is exactly a per-lane 8B global load, and B's
// layout is the matching 8B LDS read of t -> zero data shuffling.
// The matrix pipe absorbs the inner-product FMAs; VALU does only sigmoids.
// t[b][:] is staged to LDS with the Tensor Data Mover.
// Cross-wave reduction is fixed-order in LDS -> bitwise deterministic.
// ---------------------------------------------------------------------------
__global__ __launch_bounds__(256) void score_kernel(
    const float* __restrict__ edge, const float* __restrict__ t,
    float* __restrict__ scores) {
  __shared__ float tsh[N_];            // 16 KB
  __shared__ float part[8][16];        // per-wave partial scores
  const int wave = threadIdx.x >> 5;
  const int lane = threadIdx.x & 31;
  const int b    = blockIdx.x >> 8;            // 256 blocks per batch
  const int row0 = (blockIdx.x & 255) << 4;    // 16 rows per block
  const float* __restrict__ tb = t + (size_t)b * N_;

#ifdef USE_TDM
  if (threadIdx.x < 32) {
    // Tensor DMA: 1D tile of 4096 f32 (16 KB) global -> LDS.
    const unsigned long long ga = (unsigned long long)(const void*)tb;
    const unsigned ldsoff = (unsigned)(unsigned long long)(void*)&tsh[0];
    // D# group0: count=1 | lds_addr | global_addr[56:0] | type=2
    v4u g0 = { 1u, ldsoff, (unsigned)ga,
               (unsigned)((ga >> 32) & 0x1FFFFFFull) | (2u << 30) };
    // D# group1: data_size=4B; tensor_dim0=4096; tensor_dim1=1;
    //            tile_dim0=4096; tile_dim1=1; dim0_stride=4096
    v8i g1 = { (int)(2u << 16),
               (int)(((unsigned)N_ & 0xFFFFu) << 16),
               (int)(1u << 16),
               (int)((unsigned)N_ << 16),
               1,
               N_,
               (int)(((unsigned)N_ & 0xFFFFu) << 16),
               0 };
    v4i g2 = { 0, 0, 0, 0 };
    v4i g3 = { 0, 0, 0, 0 };
    v8i g4 = { 0, 0, 0, 0, 0, 0, 0, 0 };
    __builtin_amdgcn_tensor_load_to_lds(g0, g1, g2, g3, g4, 0);
    __builtin_amdgcn_s_wait_tensorcnt(0);
  }
  __syncthreads();
#else
  for (int i = threadIdx.x; i < N_; i += 256) tsh[i] = tb[i];
  __syncthreads();
#endif

  // Per-lane operand addressing for the WMMA f32 16x16x4 layout.
  const int arow  = lane & 15;                 // A/M row for this lane
  const int kpair = (lane >> 4) << 1;          // K offset: 0 (lanes 0-15) / 2 (lanes 16-31)
  const int kbase = wave * (N_ / 8) + kpair;   // this wave's 512-wide K chunk
  const v2f* __restrict__ e2 = (const v2f*)(
      edge + ((size_t)b * N_ + row0 + arow) * (size_t)N_ + kbase);
  const v2f* t2 = (const v2f*)(tsh + kbase);

  v8f d = {};                                  // C/D accumulator tile
#pragma unroll 4
  for (int it = 0; it < (N_ / 8) / 4; ++it) {  // 128 iters x K=4
    const v2f e  = e2[it * 2];                 // global_load_b64 (A operand)
    const v2f tv = t2[it * 2];                 // ds_load_b64     (B operand)
    v2f a;
    a.x = sigmoidf_fast(e.x);
    a.y = sigmoidf_fast(e.y);
    // D = A(16x4) * B(4x16) + D ; exact f32 MACs on the matrix pipe
    d = __builtin_amdgcn_wmma_f32_16x16x4_f32(
        false, a, false, tv, (short)0, d, false, false);
  }

  // Column N=0 of D: lane 0 holds rows 0..7 (vgpr r -> M=r),
  //                  lane 16 holds rows 8..15 (vgpr r -> M=8+r).
  if (lane == 0) {
#pragma unroll
    for (int r = 0; r < 8; ++r) part[wave][r] = d[r];
  } else if (lane == 16) {
#pragma unroll
    for (int r = 0; r < 8; ++r) part[wave][8 + r] = d[r];
  }
  __syncthreads();

  // Fixed-order cross-wave reduction: deterministic.
  if (threadIdx.x < 16) {
    float s = 0.f;
#pragma unroll
    for (int w = 0; w < 8; ++w) s += part[w][threadIdx.x];
    scores[(size_t)b * N_ + row0 + threadIdx.x] = s;
  }
}

// ---------------------------------------------------------------------------
// K3: per patch top-1 (first-index tie-break like lax.top_k), gather, scale,
// and cls-token passthrough. One thread per output element.
// ---------------------------------------------------------------------------
__global__ __launch_bounds__(256) void pool_kernel(
    const float* __restrict__ x, const float* __restrict__ scores,
    float* __restrict__ out) {
  const int i = blockIdx.x * 256 + threadIdx.x;
  const int total = B_ * OUTROWS * C_;
  if (i >= total) return;
  const int c = i % C_;
  const int r = (i / C_) % OUTROWS;
  const int b = i / (C_ * OUTROWS);
  if (r == 0) {
    out[i] = x[(size_t)b * (N_ + 1) * C_ + c];     // cls token row
  } else {
    const int p = r - 1;
    const float* s = scores + (size_t)b * N_ + p * PS;
    float best = s[0];
    int bi = 0;
#pragma unroll
    for (int j = 1; j < PS; ++j) {
      const float v = s[j];
      if (v > best) { best = v; bi = j; }          // strict > == first-occurrence tie-break
    }
    const float g = x[((size_t)b * (N_ + 1) + 1 + p * PS + bi) * C_ + c];
    out[i] = __builtin_fmaf(best, g, g);           // vals*gathered + gathered
  }
}

extern "C" void kernel_launch(void* const* d_in, const int* in_sizes, int n_in,
                              void* d_out, int out_size, void* d_ws, size_t ws_size,
                              hipStream_t stream) {
  const float* x     = (const float*)d_in[0];   // (B, N+1, C)
  const float* edge  = (const float*)d_in[1];   // (B, N, N)
  const float* theta = (const float*)d_in[2];   // (1, C)
  (void)in_sizes; (void)n_in; (void)out_size; (void)ws_size;

  float* t      = (float*)d_ws;                 // B*N floats (64 KB)
  float* scores = t + (size_t)B_ * N_;          // B*N floats (64 KB)
  float* out    = (float*)d_out;                // (B, 1025, C)

  node_gate_kernel<<<B_ * N_ / 8, 256, 0, stream>>>(x, theta, t);
  score_kernel<<<B_ * N_ / 16, 256, 0, stream>>>(edge, t, scores);
  const int total = B_ * OUTROWS * C_;
  pool_kernel<<<(total + 255) / 256, 256, 0, stream>>>(x, scores, out);
}